// Average_6459630813974
// MI455X (gfx1250) — compile-verified
//
#include <hip/hip_runtime.h>

// Fused bag-mean + classifier GEMM for MI455X (gfx1250, wave32).
// out[b][c] = (1/len_b) * sum_{r in bag b} x[r][:] . W[c][:] + bias[c]
//
// Roofline: 552 MB of x read once => ~24 us floor at 23.3 TB/s; GEMM is
// 1.46 GFLOP (negligible). Matrix op in full f32 via V_WMMA_F32_16X16X4_F32.
//
// Steady-state inner loop (verified in round-2 asm): s_clause'd
// global_load_b64 x4 (B fragments, branchless), ds_load_2addr_b64 x2
// (A fragments, conflict-free LDS stride 698), 4x v_wmma_f32_16x16x4_f32
// with overlapped partial waits.
//
// vs round 2: writeback guard hoisted to one wave-uniform scalar branch
// (bag0+16 <= n_bags); full tiles (all of them, since 20000 = 16*1250) do 8
// unguarded global_store_b32 from one base pointer with immediate offsets.

typedef __attribute__((ext_vector_type(2))) float v2f;
typedef __attribute__((ext_vector_type(8))) float v8f;

#define REL_DIM    690
#define KMAIN      688          // 172 * 4 branchless K-steps; step 173 peeled
#define LDS_STRIDE 698          // 698 % 64 == 58 -> conflict-free, 8B-aligned
#define N_CLASSES  53

__global__ __launch_bounds__(128) void bag_mean_wmma_kernel(
    const float* __restrict__ x,      // [n_sent, REL_DIM]
    const int*   __restrict__ scope,  // [n_bags + 1]
    const float* __restrict__ W,      // [N_CLASSES, REL_DIM]
    const float* __restrict__ bias,   // [N_CLASSES]
    float*       __restrict__ out,    // [n_bags, N_CLASSES]
    int n_bags)
{
    __shared__ float means[16 * LDS_STRIDE];   // ~43.7 KB of 320 KB WGP LDS

    const int tid  = threadIdx.x;
    const int wave = tid >> 5;                 // 0..3
    const int lane = tid & 31;
    const int bag0 = blockIdx.x * 16;

    // ---------------- Phase 1: per-bag means into LDS ----------------
    // Wave w computes tile rows 4w..4w+3. Lane owns float2 chunks
    // c2 = lane + 32*j (345 chunks cover 690 columns exactly); b64 loads
    // are coalesced across the wave (row stride 2760 B is 8B-aligned).
    for (int i = 0; i < 4; ++i) {
        const int m   = wave * 4 + i;          // tile row 0..15
        const int bag = bag0 + m;
        int   rbeg = 0, rend = 0;
        float inv  = 0.0f;
        if (bag < n_bags) {
            rbeg = scope[bag];
            rend = scope[bag + 1];
            inv  = 1.0f / (float)(rend - rbeg);   // bags are non-empty
        }
        const float2* xrow0 = (const float2*)(x + (size_t)rbeg * REL_DIM);
        for (int c2 = lane; c2 < REL_DIM / 2; c2 += 32) {
            float sx = 0.0f, sy = 0.0f;
            const float2* p = xrow0 + c2;
            for (int r = rbeg; r < rend; ++r, p += REL_DIM / 2) {
                float2 v = *p;
                sx += v.x;
                sy += v.y;
            }
            float2 mv; mv.x = sx * inv; mv.y = sy * inv;
            *(float2*)&means[m * LDS_STRIDE + 2 * c2] = mv;   // 8B-aligned
        }
        // zero the K padding columns 690..697 (exact zeros for the WMMA tail)
        if (lane < LDS_STRIDE - REL_DIM) {
            means[m * LDS_STRIDE + REL_DIM + lane] = 0.0f;
        }
    }
    __syncthreads();

    // ---------------- Phase 2: WMMA f32 16x16x4 GEMM ----------------
    // Wave w owns classes [16w, 16w+16). Layouts (ISA 7.12.2):
    //   A 16x4 f32: lanes 0-15 -> M=lane, K=k4+{0,1}; lanes 16-31 -> K=k4+{2,3}
    //   B 4x16 f32: mirrored; B[k][n] = W[n][k]
    const int  half   = lane >> 4;             // 0 or 1
    const int  l16    = lane & 15;
    const int  koff   = half * 2;
    const int  cls    = wave * 16 + l16;
    const bool cls_ok = (cls < N_CLASSES);
    // Clamp invalid classes to row 0: their B columns (n >= 53 within the
    // tile) produce garbage only in D columns we never write back.
    const float* wrow = W + (size_t)(cls_ok ? cls : 0) * REL_DIM + koff;
    const float* a_pt = &means[l16 * LDS_STRIDE + koff];

    v8f acc = {};
#pragma unroll 4
    for (int k4 = 0; k4 < KMAIN; k4 += 4) {
        v2f a, b;                               // both 8B-aligned pair loads
        a.x = a_pt[k4 + 0];
        a.y = a_pt[k4 + 1];
        b.x = wrow[k4 + 0];
        b.y = wrow[k4 + 1];
        // (neg_a, A, neg_b, B, c_mod, C, reuse_a, reuse_b)
        acc = __builtin_amdgcn_wmma_f32_16x16x4_f32(
            false, a, false, b, (short)0, acc, false, false);
    }
    // Peeled final K-step (k4 = 688): K = 688..691; columns 690/691 must be
    // zero in B (out of range of the 690-wide W rows); A side is zero-padded
    // in LDS already.
    {
        v2f a, b;
        a.x = a_pt[KMAIN + 0];
        a.y = a_pt[KMAIN + 1];
        if (half == 0) {                        // K = 688, 689: in range
            b.x = wrow[KMAIN + 0];
            b.y = wrow[KMAIN + 1];
        } else {                                // K = 690, 691: zero pad
            b.x = 0.0f;
            b.y = 0.0f;
        }
        acc = __builtin_amdgcn_wmma_f32_16x16x4_f32(
            false, a, false, b, (short)0, acc, false, false);
    }

    // -------- Writeback: D VGPR r, lane l -> M = r + 8*half, N = l16 --------
    // Lane l writes 8 outputs: out[bag0 + r + 8*half][cls], r = 0..7.
    if (cls_ok) {
        const float bv = bias[cls];
        float* op = out + (size_t)(bag0 + 8 * half) * N_CLASSES + cls;
        if (bag0 + 16 <= n_bags) {
            // Full tile (always true for n_bags % 16 == 0): unguarded stores,
            // one base pointer + immediate offsets (row stride 212 B).
#pragma unroll
            for (int r = 0; r < 8; ++r) {
                op[r * N_CLASSES] = acc[r] + bv;
            }
        } else {
#pragma unroll
            for (int r = 0; r < 8; ++r) {
                if (bag0 + r + 8 * half < n_bags) {
                    op[r * N_CLASSES] = acc[r] + bv;
                }
            }
        }
    }
}

extern "C" void kernel_launch(void* const* d_in, const int* in_sizes, int n_in,
                              void* d_out, int out_size, void* d_ws, size_t ws_size,
                              hipStream_t stream) {
    // setup_inputs() order: x (f32), scope (int), rel_weight (f32), bias (f32)
    const float* x     = (const float*)d_in[0];
    const int*   scope = (const int*)d_in[1];   // harness delivers ints as int32
    const float* W     = (const float*)d_in[2];
    const float* bias  = (const float*)d_in[3];
    float*       out   = (float*)d_out;

    const int n_bags = in_sizes[1] - 1;         // 20000
    const int blocks = (n_bags + 15) / 16;      // 1250

    bag_mean_wmma_kernel<<<dim3(blocks), dim3(128), 0, stream>>>(
        x, scope, W, bias, out, n_bags);
}